// MeanAggregator_4011499454941
// MI455X (gfx1250) — compile-verified
//
#include <hip/hip_runtime.h>

#define FEAT 128
#define WAVE_SIZE 32
#define WAVES_PER_BLOCK 8
#define PIPE 16

#if __has_builtin(__builtin_amdgcn_global_load_async_to_lds_b128)
#define USE_ASYNC_LDS 1
#else
#define USE_ASYNC_LDS 0
#endif

// Match the builtin's parameter pointee type exactly (per hipcc diagnostic):
//   __attribute__((__vector_size__(4 * sizeof(int)))) int
typedef int amd_v4i __attribute__((vector_size(4 * sizeof(int))));
typedef __attribute__((address_space(1))) amd_v4i* global_v4i_ptr;
typedef __attribute__((address_space(3))) amd_v4i* lds_v4i_ptr;

__device__ __forceinline__ void wait_asynccnt0() {
#if __has_builtin(__builtin_amdgcn_s_wait_asynccnt)
  __builtin_amdgcn_s_wait_asynccnt(0);
#else
  asm volatile("s_wait_asynccnt 0" ::: "memory");
#endif
}

// Copy 16 bytes per lane (one full 512B feature row per wave) from global to LDS.
__device__ __forceinline__ void row_copy16(const float* __restrict__ g, float* l) {
#if USE_ASYNC_LDS
  __builtin_amdgcn_global_load_async_to_lds_b128(
      (global_v4i_ptr)g,
      (lds_v4i_ptr)l,
      /*offset=*/0, /*cpol=*/0);
#else
  *reinterpret_cast<float4*>(l) = *reinterpret_cast<const float4*>(g);
#endif
}

// Shared inner loop: async-gather rows [lo,hi) of `features` (512B each),
// mean-reduce, write out row `seg`. `lrow` = this wave's PIPE-slot LDS buffer.
__device__ __forceinline__ void aggregate_segment(const float* __restrict__ features,
                                                  const int* __restrict__ neigh_idx,
                                                  float* __restrict__ out,
                                                  int seg, int lo, int hi,
                                                  int lane, float* lrow) {
  const int count = hi - lo;
  const int c4 = lane * 4;  // this lane's feature columns [c4, c4+4)
  float a0 = 0.f, a1 = 0.f, a2 = 0.f, a3 = 0.f;

  int base = lo;
  // Steady state: 16 async 512B row-gathers in flight per wave (ASYNCcnt).
  for (; base + PIPE <= hi; base += PIPE) {
    int rows[PIPE];
#pragma unroll
    for (int k = 0; k < PIPE; ++k) rows[k] = neigh_idx[base + k];
#pragma unroll
    for (int k = 0; k < PIPE; ++k)
      row_copy16(features + (size_t)rows[k] * FEAT + c4, lrow + k * FEAT + c4);
    wait_asynccnt0();
#pragma unroll
    for (int k = 0; k < PIPE; ++k) {
      const float4 v = *reinterpret_cast<const float4*>(lrow + k * FEAT + c4);
      a0 += v.x; a1 += v.y; a2 += v.z; a3 += v.w;
    }
  }
  // Tail chunk (uniform trip count within the wave).
  const int m = hi - base;
  if (m > 0) {
    for (int k = 0; k < m; ++k)
      row_copy16(features + (size_t)neigh_idx[base + k] * FEAT + c4,
                 lrow + k * FEAT + c4);
    wait_asynccnt0();
    for (int k = 0; k < m; ++k) {
      const float4 v = *reinterpret_cast<const float4*>(lrow + k * FEAT + c4);
      a0 += v.x; a1 += v.y; a2 += v.z; a3 += v.w;
    }
  }

  const float inv = 1.0f / (float)(count > 0 ? count : 1);
  float4 r;
  r.x = a0 * inv; r.y = a1 * inv; r.z = a2 * inv; r.w = a3 * inv;
  *reinterpret_cast<float4*>(out + (size_t)seg * FEAT + c4) = r;
}

// ---------------------------------------------------------------------------
// Kernel A: one pass over sorted seg_ids -> seg_start[0..B] boundary table.
// seg_start[s] = first edge of segment s; empty segments get the next edge's
// position, so count = seg_start[s+1] - seg_start[s] == 0 for them.
// ---------------------------------------------------------------------------
__global__ __launch_bounds__(256)
void seg_bounds_kernel(const int* __restrict__ seg_ids, int* __restrict__ seg_start,
                       int num_edges, int num_batch) {
  const int e = blockIdx.x * blockDim.x + threadIdx.x;
  if (e >= num_edges) return;
  const int s = seg_ids[e];
  const int sprev = (e == 0) ? -1 : seg_ids[e - 1];
  for (int j = sprev + 1; j <= s; ++j) seg_start[j] = e;
  if (e == num_edges - 1) {
    for (int j = s + 1; j <= num_batch; ++j) seg_start[j] = num_edges;
  }
}

// ---------------------------------------------------------------------------
// Kernel B (main): one wave per destination segment; range from seg_start.
// ---------------------------------------------------------------------------
__global__ __launch_bounds__(WAVES_PER_BLOCK * WAVE_SIZE)
void seg_mean_from_bounds(const float* __restrict__ features,
                          const int* __restrict__ neigh_idx,
                          const int* __restrict__ seg_start,
                          float* __restrict__ out,
                          int num_batch) {
  __shared__ float lbuf[WAVES_PER_BLOCK][PIPE][FEAT];
  const int lane = threadIdx.x & (WAVE_SIZE - 1);
  const int wave = threadIdx.x >> 5;
  const int seg  = blockIdx.x * WAVES_PER_BLOCK + wave;
  if (seg >= num_batch) return;  // uniform per wave

  const int lo = seg_start[seg];
  const int hi = seg_start[seg + 1];
  aggregate_segment(features, neigh_idx, out, seg, lo, hi, lane,
                    &lbuf[wave][0][0]);
}

// ---------------------------------------------------------------------------
// Fallback: per-wave binary search (used only if d_ws is too small).
// ---------------------------------------------------------------------------
__device__ __forceinline__ int lower_bound_i32(const int* __restrict__ a, int n, int key) {
  int lo = 0, hi = n;
  while (lo < hi) {
    int mid = (lo + hi) >> 1;
    if (a[mid] < key) lo = mid + 1; else hi = mid;
  }
  return lo;
}

__global__ __launch_bounds__(WAVES_PER_BLOCK * WAVE_SIZE)
void seg_mean_search(const float* __restrict__ features,
                     const int* __restrict__ neigh_idx,
                     const int* __restrict__ seg_ids,
                     float* __restrict__ out,
                     int num_edges, int num_batch) {
  __shared__ float lbuf[WAVES_PER_BLOCK][PIPE][FEAT];
  const int lane = threadIdx.x & (WAVE_SIZE - 1);
  const int wave = threadIdx.x >> 5;
  const int seg  = blockIdx.x * WAVES_PER_BLOCK + wave;
  if (seg >= num_batch) return;

  const int lo = lower_bound_i32(seg_ids, num_edges, seg);
  const int hi = lower_bound_i32(seg_ids, num_edges, seg + 1);
  aggregate_segment(features, neigh_idx, out, seg, lo, hi, lane,
                    &lbuf[wave][0][0]);
}

extern "C" void kernel_launch(void* const* d_in, const int* in_sizes, int n_in,
                              void* d_out, int out_size, void* d_ws, size_t ws_size,
                              hipStream_t stream) {
  (void)n_in;
  const float* features = (const float*)d_in[0];
  const int*   neigh    = (const int*)d_in[1];
  const int*   segs     = (const int*)d_in[2];
  const int E = in_sizes[1];
  const int B = out_size / FEAT;
  float* out = (float*)d_out;

  dim3 block(WAVES_PER_BLOCK * WAVE_SIZE);
  dim3 grid((B + WAVES_PER_BLOCK - 1) / WAVES_PER_BLOCK);

  if (d_ws != nullptr && ws_size >= (size_t)(B + 1) * sizeof(int)) {
    int* seg_start = (int*)d_ws;
    seg_bounds_kernel<<<dim3((E + 255) / 256), dim3(256), 0, stream>>>(
        segs, seg_start, E, B);
    seg_mean_from_bounds<<<grid, block, 0, stream>>>(
        features, neigh, seg_start, out, B);
  } else {
    seg_mean_search<<<grid, block, 0, stream>>>(
        features, neigh, segs, out, E, B);
  }
}